// MoELoRALinearLayer_53867479826599
// MI455X (gfx1250) — compile-verified
//
#include <hip/hip_runtime.h>
#include <math.h>

// Problem constants (from reference): B=4, S=4096 -> T=16384 tokens
#define T_TOKENS 16384
#define HDIM     2048
#define RANKD    128
#define NEXP     8
// ALPHA / RANK == 128/128 == 1.0f -> no extra scale needed.

typedef __attribute__((ext_vector_type(2))) float v2f;
typedef __attribute__((ext_vector_type(8))) float v8f;

// V_WMMA_F32_16X16X4_F32 : D(16x16) = A(16x4) * B(4x16) + C
// A layout (ISA 7.12.2): lane L holds row M=L%16; VGPR j holds K = 2*(L/16)+j
// B layout (symmetric):  lane L holds col N=L%16; VGPR j holds K = 2*(L/16)+j
// C/D layout: lane L col N=L%16, VGPR v row M = v + 8*(L/16)
__device__ __forceinline__ v8f wmma_f32(v2f a, v2f b, v8f c) {
  return __builtin_amdgcn_wmma_f32_16x16x4_f32(false, a, false, b, (short)0, c,
                                               false, false);
}

// CDNA5 async copy: global memory -> LDS, 16B per lane, tracked by ASYNCcnt.
// LDS address is taken from the vdst VGPR; low 32 bits of a flat __shared__ pointer are
// the LDS byte address (aperture truncation, ISA 10.2).
__device__ __forceinline__ void async_g2l_b128(unsigned lds_addr,
                                               const void* gaddr) {
  asm volatile("global_load_async_to_lds_b128 %0, %1, off"
               :
               : "v"(lds_addr), "v"((unsigned long long)(size_t)gaddr)
               : "memory");
}
__device__ __forceinline__ void wait_asynccnt0() {
  asm volatile("s_wait_asynccnt 0x0" ::: "memory");
}

// ---------------------------------------------------------------------------
// Kernel 1: routing. One wave = 16 tokens. B operand packs Wg rows (N=0..7)
// and Wn rows (N=8..15) so one WMMA accumulator gives logits AND noise-logits.
// ---------------------------------------------------------------------------
#define XCH 132  // 128-col K-chunk + 4 pad (bank step 4 -> conflict-free)

extern "C" __global__ __launch_bounds__(256, 1)
void moe_route(const float* __restrict__ x, const float* __restrict__ noise,
               const float* __restrict__ Wg, const float* __restrict__ Wn,
               float* __restrict__ rlo, float* __restrict__ wdense) {
  __shared__ float xs[8][16 * XCH];  // 67.5 KB
  const int tid  = threadIdx.x;
  const int lane = tid & 31;
  const int wave = tid >> 5;
  const int half = lane >> 4;   // which K-half this lane supplies
  const int l16  = lane & 15;
  const int t0   = (blockIdx.x * 8 + wave) * 16;
  const unsigned lwave = (unsigned)(size_t)(&xs[wave][0]);

  // B row pointer: cols 0..7 -> Wg, cols 8..15 -> Wn  (L2-resident, 64KB each)
  const float* Wrow =
      (l16 < 8) ? (Wg + (size_t)l16 * HDIM) : (Wn + (size_t)(l16 - 8) * HDIM);
  const float* wb = Wrow + 2 * half;

  v8f c0 = {}, c1 = {};
  for (int kc = 0; kc < HDIM; kc += 128) {
    // async-stage 16x128 x-chunk for this wave (one full row of 128 per iter)
#pragma unroll
    for (int i = 0; i < 16; ++i) {
      int f = (i * 32 + lane) * 4;
      int row = f >> 7, col = f & 127;
      async_g2l_b128(lwave + (unsigned)((row * XCH + col) * 4),
                     x + (size_t)(t0 + row) * HDIM + kc + col);
    }
    wait_asynccnt0();
    __syncthreads();
    const float* xr = &xs[wave][l16 * XCH + 2 * half];
#pragma unroll
    for (int k = 0; k < 128; k += 8) {
      v2f a0 = *(const v2f*)(xr + k);
      v2f b0 = *(const v2f*)(wb + kc + k);
      c0 = wmma_f32(a0, b0, c0);
      v2f a1 = *(const v2f*)(xr + k + 4);
      v2f b1 = *(const v2f*)(wb + kc + k + 4);
      c1 = wmma_f32(a1, b1, c1);
    }
    __syncthreads();
  }
  v8f c = c0 + c1;

  // transpose accumulator through LDS so one lane owns one token's 16 values
  float* cs = &xs[wave][0];
#pragma unroll
  for (int v = 0; v < 8; ++v) {
    int M = v + 8 * half;
    cs[M * 17 + l16] = c[v];
  }
  __syncthreads();

  if (lane < 16) {
    int t = t0 + lane;
    float g[16];
#pragma unroll
    for (int j = 0; j < 16; ++j) g[j] = cs[lane * 17 + j];
    float rl[8], p[8];
    float m = -3.0e38f;
#pragma unroll
    for (int e = 0; e < 8; ++e) {
      float nl = g[8 + e];
      float sp = (nl > 30.f) ? nl : log1pf(expf(nl));  // softplus
      rl[e] = g[e] + noise[(size_t)t * 8 + e] * sp;
      m = fmaxf(m, rl[e]);
    }
    float s = 0.f;
#pragma unroll
    for (int e = 0; e < 8; ++e) { p[e] = expf(rl[e] - m); s += p[e]; }
    // top-2 (lowest index wins ties, matching jax.lax.top_k)
    int i1 = 0;
#pragma unroll
    for (int e = 1; e < 8; ++e) if (p[e] > p[i1]) i1 = e;
    int i2 = (i1 == 0) ? 1 : 0;
#pragma unroll
    for (int e = 0; e < 8; ++e) if (e != i1 && p[e] > p[i2]) i2 = e;
    float inv = 1.f / (p[i1] + p[i2]);  // softmax denom cancels in the ratio
#pragma unroll
    for (int e = 0; e < 8; ++e) {
      rlo[(size_t)t * 8 + e] = rl[e];
      float w = (e == i1) ? p[i1] * inv : ((e == i2) ? p[i2] * inv : 0.f);
      wdense[(size_t)t * 8 + e] = w;
    }
  }
}

// ---------------------------------------------------------------------------
// Kernel 2: grouped LoRA. One block = 16 tokens, 8 waves.
//   down[16x128] = x[16x2048] @ Wd[e]^T   (each wave: one 16-wide r-chunk)
//   scale rows by w[t,e], stash in LDS
//   out[16x2048] += down_w[16x128] @ Wu[e]^T  (each wave: 256-col slice)
// ---------------------------------------------------------------------------
#define XS  (HDIM + 4)    // 2052: bank step 4 across rows
#define DSS (RANKD + 4)   // 132

extern "C" __global__ __launch_bounds__(256, 1)
void moe_main(const float* __restrict__ x, const float* __restrict__ Wd,
              const float* __restrict__ Wu, const float* __restrict__ wdense,
              float* __restrict__ out) {
  __shared__ float xs[16 * XS];    // 131328 B
  __shared__ float dls[16 * DSS];  // 8448 B
  __shared__ float wsh[16 * NEXP];
  __shared__ float esum[NEXP];
  const int tid  = threadIdx.x;
  const int lane = tid & 31;
  const int wave = tid >> 5;
  const int half = lane >> 4;
  const int l16  = lane & 15;
  const int t0   = blockIdx.x * 16;

  // async-stage x tile (16x2048) straight into LDS, 16B per lane per issue
  {
    const unsigned lbase = (unsigned)(size_t)(&xs[0]);
#pragma unroll
    for (int i = 0; i < 32; ++i) {
      int f = (tid + i * 256) * 4;
      int row = f >> 11, col = f & 2047;
      async_g2l_b128(lbase + (unsigned)((row * XS + col) * 4),
                     x + (size_t)(t0 + row) * HDIM + col);
    }
  }
  if (tid < 16 * NEXP) wsh[tid] = wdense[(size_t)t0 * NEXP + tid];
  wait_asynccnt0();
  __syncthreads();
  if (tid < NEXP) {
    float s = 0.f;
    for (int t = 0; t < 16; ++t) s += wsh[t * NEXP + tid];
    esum[tid] = s;
  }
  __syncthreads();

  v8f acc[16];
  v8f vzero = {};
#pragma unroll
  for (int i = 0; i < 16; ++i) acc[i] = vzero;

  const float* xrow = &xs[l16 * XS + 2 * half];
  const float* drow = &dls[l16 * DSS + 2 * half];

  for (int e = 0; e < NEXP; ++e) {
    if (esum[e] > 0.f) {  // uniform branch: esum is in LDS, same for all lanes
      // ---- DOWN: this wave computes down[16 tokens x 16 r] over K=2048 ----
      v8f d0 = vzero, d1 = vzero;
      const float* WdRow =
          Wd + ((size_t)e * RANKD + wave * 16 + l16) * HDIM + 2 * half;
#pragma unroll 8
      for (int k = 0; k < HDIM; k += 8) {
        v2f a0 = *(const v2f*)(xrow + k);
        v2f b0 = *(const v2f*)(WdRow + k);
        d0 = wmma_f32(a0, b0, d0);
        v2f a1 = *(const v2f*)(xrow + k + 4);
        v2f b1 = *(const v2f*)(WdRow + k + 4);
        d1 = wmma_f32(a1, b1, d1);
      }
      v8f d = d0 + d1;
      __syncthreads();  // previous expert's UP reads of dls must be done
#pragma unroll
      for (int v = 0; v < 8; ++v) {
        int M = v + 8 * half;  // fold routing weight into down rows here
        dls[M * DSS + wave * 16 + l16] = d[v] * wsh[M * NEXP + e];
      }
      __syncthreads();
      // ---- UP: this wave owns output columns [wave*256, wave*256+256) ----
      const float* WuE = Wu + (size_t)e * HDIM * RANKD;
      for (int nt = 0; nt < 16; ++nt) {
        const float* WuRow =
            WuE + (size_t)(wave * 256 + nt * 16 + l16) * RANKD + 2 * half;
        v8f u0 = vzero, u1 = vzero;
#pragma unroll
        for (int k = 0; k < RANKD; k += 8) {
          v2f a0 = *(const v2f*)(drow + k);
          v2f b0 = *(const v2f*)(WuRow + k);
          u0 = wmma_f32(a0, b0, u0);
          v2f a1 = *(const v2f*)(drow + k + 4);
          v2f b1 = *(const v2f*)(WuRow + k + 4);
          u1 = wmma_f32(a1, b1, u1);
        }
        acc[nt] = acc[nt] + u0 + u1;
      }
    }
  }

  // write out: two 16-float rows per VGPR store group, coalesced 64B runs
#pragma unroll
  for (int nt = 0; nt < 16; ++nt) {
    int n0 = wave * 256 + nt * 16 + l16;
#pragma unroll
    for (int v = 0; v < 8; ++v) {
      int M = v + 8 * half;
      out[(size_t)(t0 + M) * HDIM + n0] = acc[nt][v];
    }
  }
}

// ---------------------------------------------------------------------------
extern "C" void kernel_launch(void* const* d_in, const int* in_sizes, int n_in,
                              void* d_out, int out_size, void* d_ws,
                              size_t ws_size, hipStream_t stream) {
  const float* x  = (const float*)d_in[0];  // hidden_states [4,4096,2048]
  const float* nz = (const float*)d_in[1];  // noise [T,8]
  const float* Wg = (const float*)d_in[2];  // [8,2048]
  const float* Wn = (const float*)d_in[3];  // [8,2048]
  const float* Wd = (const float*)d_in[4];  // [8,128,2048]
  const float* Wu = (const float*)d_in[5];  // [8,2048,128]
  float* out = (float*)d_out;                       // [T,H] then
  float* rlo = out + (size_t)T_TOKENS * HDIM;       // router_logits [T,8]
  float* wdense = (float*)d_ws;                     // [T,8] dense weights

  moe_route<<<T_TOKENS / (16 * 8), 256, 0, stream>>>(x, nz, Wg, Wn, rlo, wdense);
  moe_main<<<T_TOKENS / 16, 256, 0, stream>>>(x, Wd, Wu, wdense, out);
}